// CoreRNNFW_78632261256149
// MI455X (gfx1250) — compile-verified
//
#include <hip/hip_runtime.h>
#include <hip/hip_bf16.h>

#define LAM_DECAY 0.95f
#define ETA_COEF  0.5f
#define LN_EPSF   1e-5f

#define BATCH 128
#define D_G   512
#define D_H   1024

typedef __attribute__((ext_vector_type(2))) float v2f;
typedef __attribute__((ext_vector_type(4))) float v4f;
typedef __attribute__((ext_vector_type(8))) float v8f;

// ---------------------------------------------------------------------------
// fp32 WMMA wrapper: D = A(16x4) * B(4x16) + C(16x16), exact fp32.
// ---------------------------------------------------------------------------
static __device__ __forceinline__ v8f wmma_f32_16x16x4(v2f a, v2f b, v8f c) {
#if defined(__HIP_DEVICE_COMPILE__) && __has_builtin(__builtin_amdgcn_wmma_f32_16x16x4_f32)
    return __builtin_amdgcn_wmma_f32_16x16x4_f32(false, a, false, b, (short)0, c,
                                                 false, false);
#else
    // placeholder so host pass / missing-builtin still compiles
    c[0] += a[0] * b[0];
    return c;
#endif
}

// ---------------------------------------------------------------------------
// Kernel 1: base = h_prev @ W_h^T + z @ W_g^T + b_h     (128 x 1024, fp32)
// One wave per 16x16 output tile. ISA 7.12.2 fragment layouts:
//   A 16x4 : lane<16 -> M=lane, K={0,1}; lane>=16 -> M=lane-16, K={2,3}
//   B 4x16 : VGPR v -> K = v + 2*(lane>>4), N = lane&15
//   C/D    : VGPR v -> M = v + 8*(lane>>4), N = lane&15
// Both W accesses are along the K-contiguous axis (C[i,j]=sum_k X[i,k]W[j,k]),
// so every fragment element pair is one float2 load.
// ---------------------------------------------------------------------------
__global__ __launch_bounds__(32) void base_gemm_wmma(
    const float* __restrict__ z, const float* __restrict__ hprev,
    const float* __restrict__ Wh, const float* __restrict__ Wg,
    const float* __restrict__ bh, float* __restrict__ base)
{
    const int lane = threadIdx.x;
    const int nt   = blockIdx.x;           // 64 tiles over D_H
    const int mt   = blockIdx.y;           // 8 tiles over B
    const int half = lane >> 4;            // 0 | 1
    const int l16  = lane & 15;
    const int kb   = half * 2;             // K sub-offset {0,2}
    const int mrow = mt * 16 + l16;        // A-fragment row
    const int ncol = nt * 16 + l16;        // B-fragment column

    v8f acc = {};

    // h_prev @ W_h^T  (K = 1024)
    {
        const float* __restrict__ ap = hprev + (size_t)mrow * D_H;
        const float* __restrict__ bp = Wh    + (size_t)ncol * D_H;
        for (int k0 = 0; k0 < D_H; k0 += 4) {
            v2f a = *(const v2f*)(ap + k0 + kb);
            v2f b = *(const v2f*)(bp + k0 + kb);
            acc = wmma_f32_16x16x4(a, b, acc);
        }
    }
    // z @ W_g^T  (K = 512)
    {
        const float* __restrict__ ap = z  + (size_t)mrow * D_G;
        const float* __restrict__ bp = Wg + (size_t)ncol * D_G;
        for (int k0 = 0; k0 < D_G; k0 += 4) {
            v2f a = *(const v2f*)(ap + k0 + kb);
            v2f b = *(const v2f*)(bp + k0 + kb);
            acc = wmma_f32_16x16x4(a, b, acc);
        }
    }

    const float bias  = bh[ncol];
    const int   rbase = mt * 16 + half * 8;
    #pragma unroll
    for (int v = 0; v < 8; ++v)
        base[(size_t)(rbase + v) * D_H + ncol] = acc[v] + bias;
}

// ---------------------------------------------------------------------------
// Block-wide LayerNorm (biased var, eps=1e-5) + ReLU over 1024 elements.
// x is this thread's element; result written to s_h[tid].
// ---------------------------------------------------------------------------
static __device__ __forceinline__ void ln_relu_block(
    float x, int tid, int lane, int wave, float g, float be,
    float* s_red, float* s_stat, float* s_h)
{
    float s = x, s2 = x * x;
    #pragma unroll
    for (int off = 16; off > 0; off >>= 1) {
        s  += __shfl_xor(s,  off, 32);
        s2 += __shfl_xor(s2, off, 32);
    }
    if (lane == 0) { s_red[wave] = s; s_red[32 + wave] = s2; }
    __syncthreads();
    if (wave == 0) {
        float ts = s_red[lane], ts2 = s_red[32 + lane];
        #pragma unroll
        for (int off = 16; off > 0; off >>= 1) {
            ts  += __shfl_xor(ts,  off, 32);
            ts2 += __shfl_xor(ts2, off, 32);
        }
        if (lane == 0) {
            const float mu  = ts * (1.0f / (float)D_H);
            const float var = ts2 * (1.0f / (float)D_H) - mu * mu;
            s_stat[0] = mu;
            s_stat[1] = rsqrtf(var + LN_EPSF);
        }
    }
    __syncthreads();
    const float hn = (x - s_stat[0]) * s_stat[1] * g + be;
    s_h[tid] = fmaxf(hn, 0.0f);
    __syncthreads();
}

// ---------------------------------------------------------------------------
// Kernel 2: per-batch fused fast-weight loop + Hebbian update.
// One block per batch element b; A[b] (4 MB) stays L2-resident across the
// 3 matvec passes. Final pass: NT load (last use) + NT store (never re-read).
// ---------------------------------------------------------------------------
__global__ __launch_bounds__(1024) void fastweight_step(
    const float* __restrict__ base, const float* __restrict__ A,
    const float* __restrict__ gamma, const float* __restrict__ beta,
    float* __restrict__ h_out, float* __restrict__ A_out)
{
    const int b    = blockIdx.x;
    const int tid  = threadIdx.x;
    const int lane = tid & 31;
    const int wave = tid >> 5;

    __shared__ __align__(16) float s_h[D_H];
    __shared__ float s_base[D_H];
    __shared__ float s_y[D_H];
    __shared__ float s_red[64];
    __shared__ float s_stat[2];

    const float* __restrict__ Ab = A + (size_t)b * D_H * D_H;
    const float g  = gamma[tid];
    const float be = beta[tid];

    const float xb = base[(size_t)b * D_H + tid];
    s_base[tid] = xb;
    __syncthreads();

    // h_0 = relu(LN(base))
    ln_relu_block(xb, tid, lane, wave, g, be, s_red, s_stat, s_h);

    // 3 inner fast-weight steps: h <- relu(LN(base + A h))
    for (int s = 0; s < 3; ++s) {
        for (int row = wave; row < D_H; row += 32) {
            const v4f* __restrict__ Ar = (const v4f*)(Ab + (size_t)row * D_H);
            float acc = 0.0f;
            #pragma unroll
            for (int t = 0; t < 8; ++t) {           // 1024 cols / (32 lanes * 4)
                const int j4 = lane + 32 * t;
                v4f a4 = Ar[j4];
                v4f h4 = *(const v4f*)(s_h + 4 * j4);
                acc = fmaf(a4.x, h4.x, acc);
                acc = fmaf(a4.y, h4.y, acc);
                acc = fmaf(a4.z, h4.z, acc);
                acc = fmaf(a4.w, h4.w, acc);
            }
            #pragma unroll
            for (int off = 16; off > 0; off >>= 1)
                acc += __shfl_xor(acc, off, 32);
            if (lane == 0) s_y[row] = s_base[row] + acc;
        }
        __syncthreads();
        ln_relu_block(s_y[tid], tid, lane, wave, g, be, s_red, s_stat, s_h);
    }

    // outputs: h_t, then A_k = 0.95*A + 0.5*h h^T
    h_out[(size_t)b * D_H + tid] = s_h[tid];

    const v4f* __restrict__ Ain  = (const v4f*)Ab;
    v4f* __restrict__       Aout = (v4f*)(A_out + (size_t)b * D_H * D_H);
    for (int idx = tid; idx < (D_H / 4) * D_H; idx += 1024) {
        const int i  = idx >> 8;          // row (256 float4 per row)
        const int j4 = idx & 255;
        v4f a4 = __builtin_nontemporal_load(Ain + idx);
        const float hi = ETA_COEF * s_h[i];          // uniform per wave
        v4f h4 = *(const v4f*)(s_h + 4 * j4);
        v4f r;
        r.x = fmaf(hi, h4.x, LAM_DECAY * a4.x);
        r.y = fmaf(hi, h4.y, LAM_DECAY * a4.y);
        r.z = fmaf(hi, h4.z, LAM_DECAY * a4.z);
        r.w = fmaf(hi, h4.w, LAM_DECAY * a4.w);
        __builtin_nontemporal_store(r, Aout + idx);
    }
}

// ---------------------------------------------------------------------------
extern "C" void kernel_launch(void* const* d_in, const int* in_sizes, int n_in,
                              void* d_out, int out_size, void* d_ws, size_t ws_size,
                              hipStream_t stream)
{
    (void)in_sizes; (void)n_in; (void)out_size; (void)ws_size;

    const float* z     = (const float*)d_in[0];
    const float* hprev = (const float*)d_in[1];
    const float* Aprev = (const float*)d_in[2];
    const float* Wh    = (const float*)d_in[3];
    const float* Wg    = (const float*)d_in[4];
    const float* bh    = (const float*)d_in[5];
    const float* gam   = (const float*)d_in[6];
    const float* bet   = (const float*)d_in[7];

    float* base  = (float*)d_ws;                      // 128*1024 fp32 scratch
    float* h_out = (float*)d_out;                     // 128*1024
    float* A_out = (float*)d_out + (size_t)BATCH * D_H; // 128*1024*1024

    dim3 g1(D_H / 16, BATCH / 16, 1);                 // 64 x 8 tiles
    base_gemm_wmma<<<g1, 32, 0, stream>>>(z, hprev, Wh, Wg, bh, base);

    fastweight_step<<<BATCH, 1024, 0, stream>>>(base, Aprev, gam, bet, h_out, A_out);
}